// ISTACell_admm_3547642986848
// MI455X (gfx1250) — compile-verified
//
#include <hip/hip_runtime.h>
#include <math.h>

#define NTOT 4096
#define HDIM 64

typedef float v2f __attribute__((ext_vector_type(2)));
typedef float v8f __attribute__((ext_vector_type(8)));

#define SL_RS    0
#define SL_PAP   1
#define SL_RSNEW 2
#define SL_TAU   3

// ---------------- init: zero scalar slots, compute tau -----------------
__global__ void k_init0(const float* v, float* scal) {
    if (threadIdx.x == 0) {
        scal[SL_RS] = 0.f; scal[SL_PAP] = 0.f; scal[SL_RSNEW] = 0.f;
        scal[SL_TAU] = 0.4f / (1.f + expf(-v[0]));
    }
}

// ------------- build q, vec; init CG state; rs = vec.vec ---------------
__global__ void k_build(const float* inp, const float* L, const int* mask,
                        const float* thP, const float* rho,
                        float* q, float* r, float* p, float* x, float* scal) {
    __shared__ float red[256];
    int i = blockIdx.x * 256 + threadIdx.x;
    float m = (mask[i] != 0) ? 1.f : 0.f;
    float vec = rho[0] * (L[i] - thP[i]) + m * inp[i];
    q[i] = m; r[i] = vec; p[i] = vec; x[i] = 0.f;
    red[threadIdx.x] = vec * vec;
    __syncthreads();
    for (int s = 128; s > 0; s >>= 1) {
        if (threadIdx.x < s) red[threadIdx.x] += red[threadIdx.x + s];
        __syncthreads();
    }
    if (threadIdx.x == 0) atomicAdd(&scal[SL_RS], red[0]);
}

// ---------------- 64x64x64 fp32 GEMM via V_WMMA_F32_16X16X4_F32 --------
__device__ __forceinline__ float fetch64(const float* M, int r, int c, int trans) {
    return trans ? M[c * 64 + r] : M[r * 64 + c];
}

__global__ void k_gemm64(const float* A, const float* B, float* C, int tA, int tB) {
    int lane = threadIdx.x & 31;
    int wv   = threadIdx.x >> 5;        // 16 waves
    int tm   = (wv >> 2) * 16;          // C tile row base
    int tn   = (wv & 3)  * 16;          // C tile col base
    int ml   = lane & 15;               // M (and N) within tile
    int kl   = (lane >> 4) * 2;         // K pair select per ISA A/B layout
    v8f c = {};
    for (int k0 = 0; k0 < 64; k0 += 4) {
        v2f a, b;
        a.x = fetch64(A, tm + ml, k0 + kl,     tA);
        a.y = fetch64(A, tm + ml, k0 + kl + 1, tA);
        b.x = fetch64(B, k0 + kl,     tn + ml, tB);
        b.y = fetch64(B, k0 + kl + 1, tn + ml, tB);
        c = __builtin_amdgcn_wmma_f32_16x16x4_f32(false, a, false, b,
                                                  (short)0, c, false, false);
    }
    int rh = (lane >> 4) * 8;           // C layout: VGPR v -> M = v + 8*(lane/16)
    for (int vv = 0; vv < 8; ++vv)
        C[(tm + rh + vv) * 64 + (tn + ml)] = c[vv];
}

// ------- CG matvec: y = l1*D@p + l2*(X@StS) + (q+rho)*p ; pAp accum ----
__global__ void k_matvec(const float* D, const float* p, const float* q,
                         const float* StS, const float* l1p, const float* l2p,
                         const float* rhop, float* y, float* scal) {
    __shared__ float part[8];
    int lane = threadIdx.x & 31;
    int wv   = threadIdx.x >> 5;
    int row  = blockIdx.x * 8 + wv;     // 512 blocks * 8 waves = 4096 rows
    float l1 = l1p[0], l2 = l2p[0], rho = rhop[0];
    const float4* Dr = (const float4*)(D + (size_t)row * NTOT);
    const float4* p4 = (const float4*)p;
    float accD = 0.f;
    for (int j = lane; j < NTOT / 4; j += 32) {
        float4 d = Dr[j]; float4 pv = p4[j];
        accD += d.x * pv.x + d.y * pv.y + d.z * pv.z + d.w * pv.w;
    }
    int u = row & 63, h = row >> 6;     // X@StS term: row h dot StS col u
    float accS = p[h * 64 + lane]      * StS[lane * 64 + u]
               + p[h * 64 + lane + 32] * StS[(lane + 32) * 64 + u];
    float val = l1 * accD + l2 * accS;
    for (int off = 16; off > 0; off >>= 1) val += __shfl_xor(val, off);
    if (lane == 0) {
        float pn = p[row];
        float yv = val + (q[row] + rho) * pn;
        y[row] = yv;
        part[wv] = yv * pn;
    }
    __syncthreads();
    if (threadIdx.x == 0) {
        float s = 0.f;
        for (int i = 0; i < 8; ++i) s += part[i];
        atomicAdd(&scal[SL_PAP], s);
    }
}

__global__ void k_zero(float* scal, int first) {
    if (threadIdx.x == 0) {
        if (!first) scal[SL_RS] = scal[SL_RSNEW];
        scal[SL_PAP] = 0.f;
        scal[SL_RSNEW] = 0.f;
    }
}

__global__ void k_update1(float* x, float* r, const float* p, const float* y,
                          float* scal) {
    __shared__ float red[256];
    int i = blockIdx.x * 256 + threadIdx.x;
    float pap = scal[SL_PAP];
    float alpha = (pap != 0.f) ? scal[SL_RS] / pap : 0.f;
    x[i] += alpha * p[i];
    float rn = r[i] - alpha * y[i];
    r[i] = rn;
    red[threadIdx.x] = rn * rn;
    __syncthreads();
    for (int s = 128; s > 0; s >>= 1) {
        if (threadIdx.x < s) red[threadIdx.x] += red[threadIdx.x + s];
        __syncthreads();
    }
    if (threadIdx.x == 0) atomicAdd(&scal[SL_RSNEW], red[0]);
}

__global__ void k_update2(const float* r, float* p, const float* scal) {
    int i = blockIdx.x * 256 + threadIdx.x;
    float rs = scal[SL_RS];
    float beta = (rs != 0.f) ? scal[SL_RSNEW] / rs : 0.f;
    p[i] = r[i] + beta * p[i];
}

__global__ void k_makeM(const float* x, const float* thP, float* M) {
    int i = blockIdx.x * 256 + threadIdx.x;
    M[i] = x[i] + thP[i];
}

// ----- one-workgroup parallel cyclic Jacobi eigensolver on B = M^T M ---
// outputs V and W = V * diag(relu(s - tau*smax)/s)
__global__ void k_jacobi(const float* Bin, const float* scal,
                         float* Vout, float* Wout) {
    __shared__ float B[64][65];
    __shared__ float V[64][65];
    __shared__ float cs[32], sn[32];
    __shared__ int   ord[64], pi[32], qi[32];
    __shared__ float sv[64];
    __shared__ float smax;
    int t = threadIdx.x;                 // 256 threads
    for (int i = t; i < 64 * 64; i += 256) {
        int rr = i >> 6, cc = i & 63;
        B[rr][cc] = Bin[i];
        V[rr][cc] = (rr == cc) ? 1.f : 0.f;
    }
    if (t < 64) ord[t] = t;
    __syncthreads();
    for (int sweep = 0; sweep < 12; ++sweep) {
        for (int step = 0; step < 63; ++step) {
            if (t < 32) {
                int p = ord[t], q = ord[63 - t];
                if (p > q) { int tmp = p; p = q; q = tmp; }
                pi[t] = p; qi[t] = q;
            }
            __syncthreads();
            if (t < 32) {
                int p = pi[t], q = qi[t];
                float app = B[p][p], aqq = B[q][q], apq = B[p][q];
                float c = 1.f, s = 0.f;
                if (fabsf(apq) > 1e-12f) {
                    float zeta = (aqq - app) / (2.f * apq);
                    float tt = copysignf(1.f, zeta) /
                               (fabsf(zeta) + sqrtf(1.f + zeta * zeta));
                    c = rsqrtf(1.f + tt * tt);
                    s = tt * c;
                }
                cs[t] = c; sn[t] = s;
            }
            __syncthreads();
            // row update: B <- J^T B  (pairs own disjoint rows)
            for (int i = t; i < 32 * 64; i += 256) {
                int pr = i >> 6, k = i & 63;
                int p = pi[pr], q = qi[pr];
                float c = cs[pr], s = sn[pr];
                float bp = B[p][k], bq = B[q][k];
                B[p][k] = c * bp - s * bq;
                B[q][k] = s * bp + c * bq;
            }
            __syncthreads();
            // col update: B <- B J ; V <- V J
            for (int i = t; i < 32 * 64; i += 256) {
                int pr = i >> 6, k = i & 63;
                int p = pi[pr], q = qi[pr];
                float c = cs[pr], s = sn[pr];
                float bp = B[k][p], bq = B[k][q];
                B[k][p] = c * bp - s * bq;
                B[k][q] = s * bp + c * bq;
                float vp = V[k][p], vq = V[k][q];
                V[k][p] = c * vp - s * vq;
                V[k][q] = s * vp + c * vq;
            }
            __syncthreads();
            if (t == 0) {                // rotate round-robin schedule
                int last = ord[63];
                for (int i = 63; i > 1; --i) ord[i] = ord[i - 1];
                ord[1] = last;
            }
            __syncthreads();
        }
    }
    if (t < 64) sv[t] = sqrtf(fmaxf(B[t][t], 0.f));
    __syncthreads();
    if (t == 0) {
        float mx = 0.f;
        for (int i = 0; i < 64; ++i) mx = fmaxf(mx, sv[i]);
        smax = mx;
    }
    __syncthreads();
    float tau = scal[SL_TAU];
    for (int i = t; i < 64 * 64; i += 256) {
        int rr = i >> 6, k = i & 63;
        float s = sv[k];
        float thr = fmaxf(s - tau * smax, 0.f);
        float ratio = (s > 1e-20f) ? thr / s : 0.f;
        Vout[i] = V[rr][k];
        Wout[i] = V[rr][k] * ratio;
    }
}

__global__ void k_final(const float* Lt, const float* x, const float* thP,
                        const float* neta, float* out) {
    int i = blockIdx.x * 256 + threadIdx.x;
    out[i] = Lt[i];
    out[NTOT + i] = thP[i] + neta[0] * (x[i] - Lt[i]);
}

extern "C" void kernel_launch(void* const* d_in, const int* in_sizes, int n_in,
                              void* d_out, int out_size, void* d_ws, size_t ws_size,
                              hipStream_t stream) {
    const float* inp  = (const float*)d_in[0];
    const float* L    = (const float*)d_in[1];
    const int*   mask = (const int*)d_in[2];
    const float* D    = (const float*)d_in[3];
    const float* thP  = (const float*)d_in[4];
    const float* v    = (const float*)d_in[5];
    const float* neta = (const float*)d_in[6];
    const float* l1   = (const float*)d_in[7];
    const float* l2   = (const float*)d_in[8];
    const float* rho  = (const float*)d_in[9];
    const float* S    = (const float*)d_in[10];
    float* out = (float*)d_out;

    float* w    = (float*)d_ws;
    float* scal = w;                 // 64 floats of scalar slots
    float* q    = w + 64;
    float* r    = q  + NTOT;
    float* p    = r  + NTOT;
    float* x    = p  + NTOT;
    float* y    = x  + NTOT;
    float* StS  = y  + NTOT;
    float* M    = StS + NTOT;
    float* B2   = M  + NTOT;
    float* V    = B2 + NTOT;
    float* W    = V  + NTOT;
    float* T    = W  + NTOT;
    float* Lt   = T  + NTOT;

    k_init0<<<1, 64, 0, stream>>>(v, scal);
    k_build<<<16, 256, 0, stream>>>(inp, L, mask, thP, rho, q, r, p, x, scal);
    k_gemm64<<<1, 512, 0, stream>>>(S, S, StS, 1, 0);        // StS = S^T S (WMMA)

    for (int it = 0; it < 20; ++it) {
        k_zero<<<1, 32, 0, stream>>>(scal, it == 0 ? 1 : 0);
        k_matvec<<<512, 256, 0, stream>>>(D, p, q, StS, l1, l2, rho, y, scal);
        k_update1<<<16, 256, 0, stream>>>(x, r, p, y, scal);
        k_update2<<<16, 256, 0, stream>>>(r, p, scal);
    }

    k_makeM<<<16, 256, 0, stream>>>(x, thP, M);
    k_gemm64<<<1, 512, 0, stream>>>(M, M, B2, 1, 0);          // B = M^T M (WMMA)
    k_jacobi<<<1, 256, 0, stream>>>(B2, scal, V, W);
    k_gemm64<<<1, 512, 0, stream>>>(W, V, T, 0, 1);           // T = W V^T (WMMA)
    k_gemm64<<<1, 512, 0, stream>>>(M, T, Lt, 0, 0);          // Lt = M T (WMMA)
    k_final<<<16, 256, 0, stream>>>(Lt, x, thP, neta, out);
}